// AT_VQVAE_Encoder_84774064488650
// MI455X (gfx1250) — compile-verified
//
#include <hip/hip_runtime.h>
#include <hip/hip_bf16.h>
#include <math.h>

// ---------------------------------------------------------------------------
// Cross-modal VQ-VAE encoder forward for MI455X (gfx1250).
// Heavy op: fp32 distance matmul x@E^T via V_WMMA_F32_16X16X4_F32 (wave32).
// ---------------------------------------------------------------------------

typedef __attribute__((ext_vector_type(2))) float v2f;
typedef __attribute__((ext_vector_type(8))) float v8f;

#define NROWS   32768   // B*T
#define BDIM    64
#define TDIM    512
#define DDIM    256
#define MCODES  400
#define NTILES  25      // 400 / 16
#define ROWS    32      // rows per workgroup in the WMMA kernel
#define XST     260     // padded LDS row stride (floats)
#define DST     408     // padded dot-block stride (floats)

// ws layout (in 4-byte elements)
#define WS_COUNTS_A 0
#define WS_COUNTS_V 25600
#define WS_PH_A     51200
#define WS_PH_V     76800
#define WS_SUMS     102400
#define WS_ZEROCNT  102408
#define WS_ESQ      102408
#define WS_IDX_A    102808
#define WS_IDX_V    135576
#define WS_LA       168344
#define WS_LV       193944
#define WS_SCODE    219544

// ---------------------------------------------------------------------------
// K0: zero accumulators, precompute ||e||^2 per code
// ---------------------------------------------------------------------------
__global__ __launch_bounds__(256) void vq_init(float* __restrict__ wsf,
                                               const float* __restrict__ emb) {
  int i = blockIdx.x * 256 + threadIdx.x;
  if (i < WS_ZEROCNT) wsf[i] = 0.0f;          // counts(int 0 == f32 0), pH, sums
  if (i < MCODES) {
    float s = 0.0f;
    const float* e = emb + (size_t)i * DDIM;
    for (int k = 0; k < DDIM; ++k) s += e[k] * e[k];
    wsf[WS_ESQ + i] = s;
  }
}

// ---------------------------------------------------------------------------
// K1: WMMA distance kernel. One WG = 64 threads (2 waves), 32 rows, one batch.
//   pass1: dot[r][c] = x_r . e_c via v_wmma_f32_16x16x4_f32 (K=256 -> 64 steps)
//   pass2: per-row argmin + softmax(-sqrt(d)) row max/sum
//   pass3: per-code probability column sums -> atomic into pH[b][c]
// ---------------------------------------------------------------------------
__global__ __launch_bounds__(64) void vq_dist_kernel(
    const float* __restrict__ audio, const float* __restrict__ video,
    const float* __restrict__ emb, const float* __restrict__ e_sq,
    int* __restrict__ idx_a, int* __restrict__ idx_v,
    int* __restrict__ counts_a, int* __restrict__ counts_v,
    float* __restrict__ pH_a, float* __restrict__ pH_v) {
  const int tid  = threadIdx.x;
  const int wave = tid >> 5;
  const int lane = tid & 31;
  const int n0   = blockIdx.x * ROWS;
  const int b    = n0 / TDIM;                  // 512/32=16 blocks per batch
  const float* __restrict__ x = blockIdx.y ? video : audio;
  int*   __restrict__ idxp = blockIdx.y ? idx_v : idx_a;
  int*   __restrict__ cntp = blockIdx.y ? counts_v : counts_a;
  float* __restrict__ pHp  = blockIdx.y ? pH_v : pH_a;

  __shared__ float xs[ROWS * XST];             // 33280 B
  __shared__ float es[16 * XST];               // 16640 B
  __shared__ float dotb[ROWS * DST];           // 52224 B
  __shared__ float esq_s[MCODES];
  __shared__ float rowsq[ROWS];
  __shared__ float rowm[ROWS];
  __shared__ float rowsum[ROWS];

  // stage 32 x-rows (float4)
  for (int i = tid; i < ROWS * (DDIM / 4); i += 64) {
    int r = i >> 6, c4 = i & 63;
    float4 v = *(const float4*)&x[(size_t)(n0 + r) * DDIM + c4 * 4];
    *(float4*)&xs[r * XST + c4 * 4] = v;
  }
  for (int c = tid; c < MCODES; c += 64) esq_s[c] = e_sq[c];
  __syncthreads();

  if (tid < ROWS) {                            // row squared norms from LDS
    float s = 0.0f;
    for (int k = 0; k < DDIM; ++k) { float v = xs[tid * XST + k]; s += v * v; }
    rowsq[tid] = s;
  }

  const float* ap = &xs[(wave * 16 + (lane & 15)) * XST + 2 * (lane >> 4)];
  const float* bp = &es[(lane & 15) * XST + 2 * (lane >> 4)];

  for (int t = 0; t < NTILES; ++t) {
    __syncthreads();                           // prior tile done reading es
    for (int i = tid; i < 16 * (DDIM / 4); i += 64) {
      int r = i >> 6, c4 = i & 63;
      float4 v = *(const float4*)&emb[(size_t)(t * 16 + r) * DDIM + c4 * 4];
      *(float4*)&es[r * XST + c4 * 4] = v;
    }
    __syncthreads();

    v8f acc = {};
#pragma unroll 8
    for (int k = 0; k < DDIM; k += 4) {
      v2f a = *(const v2f*)(ap + k);
      v2f bb = *(const v2f*)(bp + k);
      acc = __builtin_amdgcn_wmma_f32_16x16x4_f32(
          false, a, false, bb, (short)0, acc, false, false);
    }
    const int code  = t * 16 + (lane & 15);
    const int rbase = wave * 16 + 8 * (lane >> 4);
#pragma unroll
    for (int i = 0; i < 8; ++i) dotb[(rbase + i) * DST + code] = acc[i];
  }
  __syncthreads();

  // pass2: per-row argmin (first-min tie rule) + softmax max & sum
  if (tid < ROWS) {
    const int r = tid;
    const float rs = rowsq[r];
    float dmin = 3.4e38f; int arg = 0;
    for (int c = 0; c < MCODES; ++c) {
      float d = rs + esq_s[c] - 2.0f * dotb[r * DST + c];
      if (d < dmin) { dmin = d; arg = c; }
    }
    float m = sqrtf(fmaxf(dmin, 0.0f));        // = -max(-sqrt(d))
    float s = 0.0f;
    for (int c = 0; c < MCODES; ++c) {
      float d = rs + esq_s[c] - 2.0f * dotb[r * DST + c];
      s += expf(m - sqrtf(fmaxf(d, 0.0f)));
    }
    rowm[r] = m; rowsum[r] = s;
    idxp[n0 + r] = arg;
    atomicAdd(&cntp[b * MCODES + arg], 1);
  }
  __syncthreads();

  // pass3: column sums of probabilities -> pH[b][c] (divide by T later)
  for (int c = tid; c < MCODES; c += 64) {
    float accp = 0.0f;
    const float ec = esq_s[c];
    for (int r = 0; r < ROWS; ++r) {
      float d = rowsq[r] + ec - 2.0f * dotb[r * DST + c];
      accp += expf(rowm[r] - sqrtf(fmaxf(d, 0.0f))) / rowsum[r];
    }
    atomicAdd(&pHp[b * MCODES + c], accp);
  }
}

// ---------------------------------------------------------------------------
// K2: gather quantized outputs (forward value of straight-through) + MSE sums
// one block per row n; 256 threads = D
// ---------------------------------------------------------------------------
__global__ __launch_bounds__(256) void vq_gather_loss(
    const float* __restrict__ audio, const float* __restrict__ video,
    const float* __restrict__ emb,
    const int* __restrict__ idx_a, const int* __restrict__ idx_v,
    float* __restrict__ out_aq, float* __restrict__ out_vq,
    float* __restrict__ sums) {
  const int n = blockIdx.x;
  const int d = threadIdx.x;
  const int ia = idx_a[n], iv = idx_v[n];
  float a  = audio[(size_t)n * DDIM + d];
  float v  = video[(size_t)n * DDIM + d];
  float aq = emb[(size_t)ia * DDIM + d];
  float vq = emb[(size_t)iv * DDIM + d];
  out_aq[(size_t)n * DDIM + d] = aq;
  out_vq[(size_t)n * DDIM + d] = vq;

  __shared__ float4 red4[256];
  float4 p;
  p.x = (a - aq) * (a - aq);   // a_e
  p.y = (a - vq) * (a - vq);   // av_e
  p.z = (v - vq) * (v - vq);   // v_e
  p.w = (v - aq) * (v - aq);   // va_e
  red4[d] = p;
  __syncthreads();
  for (int s = 128; s > 0; s >>= 1) {
    if (d < s) {
      red4[d].x += red4[d + s].x; red4[d].y += red4[d + s].y;
      red4[d].z += red4[d + s].z; red4[d].w += red4[d + s].w;
    }
    __syncthreads();
  }
  if (d == 0) {
    atomicAdd(&sums[0], red4[0].x);
    atomicAdd(&sums[1], red4[0].y);
    atomicAdd(&sums[2], red4[0].z);
    atomicAdd(&sums[3], red4[0].w);
  }
}

// ---------------------------------------------------------------------------
// K3a: pH /= T and log(pH + 1e-10)
// ---------------------------------------------------------------------------
__global__ __launch_bounds__(256) void vq_norm_log(
    float* __restrict__ pH_a, float* __restrict__ pH_v,
    float* __restrict__ la, float* __restrict__ lv) {
  int i = blockIdx.x * 256 + threadIdx.x;
  const int NM = BDIM * MCODES;
  if (i < NM) {
    float p = pH_a[i] * (1.0f / TDIM);
    pH_a[i] = p; la[i] = logf(p + 1e-10f);
  } else if (i < 2 * NM) {
    int j = i - NM;
    float p = pH_v[j] * (1.0f / TDIM);
    pH_v[j] = p; lv[j] = logf(p + 1e-10f);
  }
}

// ---------------------------------------------------------------------------
// K3b: Scode[i][j] = sum_c pHa[i,c]*lv[j,c] + pHv[i,c]*la[j,c]   (64x64)
// ---------------------------------------------------------------------------
__global__ __launch_bounds__(256) void vq_scode(
    const float* __restrict__ pH_a, const float* __restrict__ pH_v,
    const float* __restrict__ la, const float* __restrict__ lv,
    float* __restrict__ Scode) {
  int e = blockIdx.x * 256 + threadIdx.x;      // 0..4095
  int i = e >> 6, j = e & 63;
  float s = 0.0f;
  for (int c = 0; c < MCODES; ++c)
    s += pH_a[i * MCODES + c] * lv[j * MCODES + c] +
         pH_v[i * MCODES + c] * la[j * MCODES + c];
  Scode[e] = s;
}

// ---------------------------------------------------------------------------
// K3c: final scalars (one block)
// ---------------------------------------------------------------------------
__device__ float blockReduceSum(float v, float* red) {
  int tid = threadIdx.x;
  __syncthreads();
  red[tid] = v; __syncthreads();
  for (int s = 128; s > 0; s >>= 1) {
    if (tid < s) red[tid] += red[tid + s];
    __syncthreads();
  }
  return red[0];
}
__device__ float blockReduceMin(float v, float* red) {
  int tid = threadIdx.x;
  __syncthreads();
  red[tid] = v; __syncthreads();
  for (int s = 128; s > 0; s >>= 1) {
    if (tid < s) red[tid] = fminf(red[tid], red[tid + s]);
    __syncthreads();
  }
  return red[0];
}

__global__ __launch_bounds__(256) void vq_final(
    const int* __restrict__ counts_a, const int* __restrict__ counts_v,
    const float* __restrict__ sums, const float* __restrict__ Scode,
    float* __restrict__ out_sc) {
  __shared__ float red[256];
  __shared__ float scl[4096];
  __shared__ int eqcount;
  const int tid = threadIdx.x;
  for (int i = tid; i < 4096; i += 256) scl[i] = Scode[i];
  if (tid == 0) eqcount = 0;
  __syncthreads();

  // perplexities: bincount = sum over batches of per-batch counts
  float ha = 0.0f, hv = 0.0f;
  for (int c = tid; c < MCODES; c += 256) {
    int ba = 0, bv = 0;
    for (int bb = 0; bb < BDIM; ++bb) {
      ba += counts_a[bb * MCODES + c];
      bv += counts_v[bb * MCODES + c];
    }
    float pa = (float)ba * (1.0f / NROWS);
    float pv = (float)bv * (1.0f / NROWS);
    ha -= pa * logf(pa + 1e-10f);
    hv -= pv * logf(pv + 1e-10f);
  }
  float ha_t = blockReduceSum(ha, red);
  float hv_t = blockReduceSum(hv, red);

  // per-batch modes (argmax of counts, first max on ties) + agreement
  if (tid < BDIM) {
    int besta = 0, bca = counts_a[tid * MCODES];
    int bestv = 0, bcv = counts_v[tid * MCODES];
    for (int c = 1; c < MCODES; ++c) {
      int ca = counts_a[tid * MCODES + c]; if (ca > bca) { bca = ca; besta = c; }
      int cv = counts_v[tid * MCODES + c]; if (cv > bcv) { bcv = cv; bestv = c; }
    }
    if (besta == bestv) atomicAdd(&eqcount, 1);
  }

  // MaxScode = max(-Scode) = -min(Scode)
  float mn = 3.4e38f;
  for (int i = tid; i < 4096; i += 256) mn = fminf(mn, scl[i]);
  float Mx = -blockReduceMin(mn, red);

  // Lcmcm = -mean_i log(E_ii / (rowsum_i + EPS)), E = exp(Scode + Mx)
  float l = 0.0f;
  if (tid < BDIM) {
    float rs = 0.0f;
    for (int j = 0; j < BDIM; ++j) rs += expf(scl[tid * 64 + j] + Mx);
    float diag = expf(scl[tid * 65] + Mx);
    l = logf(diag / (rs + 1e-5f));
  }
  float lsum = blockReduceSum(l, red);

  if (tid == 0) {
    const float inv = 1.0f / ((float)NROWS * (float)DDIM);
    out_sc[0] = 0.5f * sums[0] * inv + 0.25f * sums[1] * inv;  // a_loss
    out_sc[1] = 0.5f * sums[2] * inv + 0.25f * sums[3] * inv;  // v_loss
    out_sc[2] = expf(ha_t);                                     // a_perplexity
    out_sc[3] = expf(hv_t);                                     // v_perplexity
    out_sc[4] = 0.5f * (-(lsum / (float)BDIM));                 // cmcm_loss
    out_sc[5] = (float)eqcount;                                 // equal_num
  }
}

// ---------------------------------------------------------------------------
extern "C" void kernel_launch(void* const* d_in, const int* in_sizes, int n_in,
                              void* d_out, int out_size, void* d_ws, size_t ws_size,
                              hipStream_t stream) {
  const float* audio = (const float*)d_in[0];
  const float* video = (const float*)d_in[1];
  const float* emb   = (const float*)d_in[2];
  float* out = (float*)d_out;
  float* wsf = (float*)d_ws;
  int*   wsi = (int*)d_ws;

  int*   counts_a = wsi + WS_COUNTS_A;
  int*   counts_v = wsi + WS_COUNTS_V;
  float* pH_a     = wsf + WS_PH_A;
  float* pH_v     = wsf + WS_PH_V;
  float* sums     = wsf + WS_SUMS;
  float* e_sq     = wsf + WS_ESQ;
  int*   idx_a    = wsi + WS_IDX_A;
  int*   idx_v    = wsi + WS_IDX_V;
  float* la       = wsf + WS_LA;
  float* lv       = wsf + WS_LV;
  float* Scode    = wsf + WS_SCODE;

  float* out_aq = out;
  float* out_vq = out + (size_t)NROWS * DDIM;
  float* out_sc = out + 2 * (size_t)NROWS * DDIM;

  vq_init<<<(WS_ZEROCNT + 255) / 256, 256, 0, stream>>>(wsf, emb);

  dim3 g1(NROWS / ROWS, 2);
  vq_dist_kernel<<<g1, 64, 0, stream>>>(audio, video, emb, e_sq,
                                        idx_a, idx_v, counts_a, counts_v,
                                        pH_a, pH_v);

  vq_gather_loss<<<NROWS, 256, 0, stream>>>(audio, video, emb, idx_a, idx_v,
                                            out_aq, out_vq, sums);

  vq_norm_log<<<(2 * BDIM * MCODES + 255) / 256, 256, 0, stream>>>(pH_a, pH_v, la, lv);
  vq_scode<<<16, 256, 0, stream>>>(pH_a, pH_v, la, lv, Scode);
  vq_final<<<1, 256, 0, stream>>>(counts_a, counts_v, sums, Scode, out_sc);
}